// NPENASGinPredictorAgent_34256659153347
// MI455X (gfx1250) — compile-verified
//
#include <hip/hip_runtime.h>
#include <hip/hip_bf16.h>

// ---------------------------------------------------------------------------
// GIN predictor for gfx1250 (MI455X).
//  - Edge aggregation: per-(edge,feature) global_atomic_add_f32, one wave per
//    edge -> 128B coalesced gather + 128B coalesced atomic (L2-resident).
//  - BatchNorm folded into consumers as a per-feature affine (no standalone
//    19MB normalize pass); affine computed by a 32-thread kernel per layer.
//  - GIN MLPs (N x 32 @ 32 x 32, twice per layer): V_WMMA_F32_16X16X4_F32,
//    exact f32. Weights staged into LDS via global_load_async_to_lds_b128
//    (ASYNCcnt path); C-layout -> A-layout restripe through LDS between the
//    two linears; BN stats reduced out of the WMMA accumulators.
// ---------------------------------------------------------------------------

typedef float v2f __attribute__((ext_vector_type(2)));
typedef float v8f __attribute__((ext_vector_type(8)));

__device__ __forceinline__ v8f wmma_f32_4(v2f a, v2f b, v8f c) {
  // D(16x16,f32) += A(16x4,f32) * B(4x16,f32), exact f32.
  return __builtin_amdgcn_wmma_f32_16x16x4_f32(
      /*neg_a=*/false, a, /*neg_b=*/false, b,
      /*c_mod=*/(short)0, c, /*reuse_a=*/false, /*reuse_b=*/false);
}

// ---------------------------------------------------------------------------
// Pad w1a (6x32) to 32x32, and write the identity affine (sc=1, sh=0).
__global__ __launch_bounds__(256) void k_pad_w1(const float* __restrict__ w1a,
                                                float* __restrict__ w1p,
                                                float* __restrict__ aff) {
  int i = blockIdx.x * 256 + threadIdx.x;
  if (i < 1024) {
    w1p[i] = (i < 6 * 32) ? w1a[i] : 0.f;
  } else if (i < 1088) {
    aff[i - 1024] = (i < 1056) ? 1.f : 0.f;  // sc[32]=1, sh[32]=0
  }
}

// Layer-1 init: XP = zero-padded x (6 -> 32 cols, zero row tail), T = XP,
// zero BN-stats accumulator.
__global__ __launch_bounds__(256) void k_init_layer1(
    const float* __restrict__ x, float* __restrict__ xp, float* __restrict__ t,
    float* __restrict__ bns, int n_valid, long total) {
  long i = (long)blockIdx.x * 256 + threadIdx.x;
  if (i < 64) bns[i] = 0.f;
  if (i >= total) return;
  int n = (int)(i >> 5), f = (int)(i & 31);
  float v = (n < n_valid && f < 6) ? x[(size_t)n * 6 + f] : 0.f;
  xp[i] = v;
  t[i] = v;
}

// Layers 2/3 init: T = affine(R) (BN applied on the fly), zero BN stats.
// Padded-tail rows get sh[f]; all downstream consumers mask rows >= N.
__global__ __launch_bounds__(256) void k_init_aff(const float* __restrict__ src,
                                                  const float* __restrict__ aff,
                                                  float* __restrict__ t,
                                                  float* __restrict__ bns,
                                                  long total) {
  long i = (long)blockIdx.x * 256 + threadIdx.x;
  if (i < 64) bns[i] = 0.f;
  if (i >= total) return;
  int f = (int)(i & 31);
  t[i] = fmaf(src[i], aff[f], aff[32 + f]);
}

// Edge scatter: T[dst] += affine(S[src]); one wave per edge (f = lane).
__global__ __launch_bounds__(256) void k_edges(const int* __restrict__ ei,
                                               const float* __restrict__ s,
                                               const float* __restrict__ aff,
                                               float* __restrict__ t, int ne) {
  long gid = (long)blockIdx.x * 256 + threadIdx.x;
  int e = (int)(gid >> 5);
  if (e >= ne) return;
  int f = (int)(gid & 31);
  int vs = ei[e];
  int vd = ei[ne + e];
  float v = fmaf(s[(size_t)vs * 32 + f], aff[f], aff[32 + f]);
  atomicAdd(&t[(size_t)vd * 32 + f], v);
}

// ---------------------------------------------------------------------------
// WMMA GIN MLP: R = relu( relu(T@W1 + B1) @ W2 + B2 ), plus per-feature
// sum / sum-of-squares accumulation for BatchNorm.
__global__ __launch_bounds__(256) void k_gin_mlp(
    const float* __restrict__ T, const float* __restrict__ W1,
    const float* __restrict__ B1, const float* __restrict__ W2,
    const float* __restrict__ B2, float* __restrict__ R,
    float* __restrict__ bns, int n_valid) {
  __shared__ __align__(16) float lds_w[2048];  // W1 (1024) | W2 (1024)
  __shared__ float lds_u[8][16 * 33];          // per-wave 16x32 U tile
  __shared__ float lds_red[8][64];             // per-wave BN partials

  const int tid = threadIdx.x;
  const int wave = tid >> 5;
  const int lane = tid & 31;
  const int half = lane >> 4;  // A/B fragments: lanes 16..31 hold K = 2*half+j
  const int l16 = lane & 15;
  const int row0 = (blockIdx.x * 8 + wave) * 16;

  // ---- stage W1|W2 into LDS via the async-to-LDS DMA path (ASYNCcnt) ----
  {
    unsigned lbase = (unsigned)(size_t)(&lds_w[0]);  // flat LDS addr[31:0]
    unsigned long long ga1 = (unsigned long long)(size_t)W1 + (unsigned)(tid * 16);
    unsigned long long ga2 = (unsigned long long)(size_t)W2 + (unsigned)(tid * 16);
    asm volatile("global_load_async_to_lds_b128 %0, %1, off"
                 :: "v"(lbase + (unsigned)(tid * 16)), "v"(ga1) : "memory");
    asm volatile("global_load_async_to_lds_b128 %0, %1, off"
                 :: "v"(lbase + 4096u + (unsigned)(tid * 16)), "v"(ga2) : "memory");
    asm volatile("s_wait_asynccnt 0x0" ::: "memory");
  }
  __syncthreads();

  // ---- GEMM1: U = relu(T@W1 + B1), K = 0..31 in steps of 4 ----
  const float* arow = T + (size_t)(row0 + l16) * 32;  // A: lane = row
  v8f c0 = {}, c1 = {};
#pragma unroll
  for (int kb = 0; kb < 32; kb += 4) {
    const int k = kb + 2 * half;
    v2f a;  a.x = arow[k];                   a.y = arow[k + 1];
    v2f bl; bl.x = lds_w[k * 32 + l16];      bl.y = lds_w[(k + 1) * 32 + l16];
    v2f bh; bh.x = lds_w[k * 32 + 16 + l16]; bh.y = lds_w[(k + 1) * 32 + 16 + l16];
    c0 = wmma_f32_4(a, bl, c0);  // cols 0..15
    c1 = wmma_f32_4(a, bh, c1);  // cols 16..31
  }
  // C layout: c[g] @ lane = U[row0 + g + 8*half][col]; restripe via LDS.
  const float b1l = B1[l16], b1h = B1[16 + l16];
#pragma unroll
  for (int g = 0; g < 8; ++g) {
    const int r = g + 8 * half;
    lds_u[wave][r * 33 + l16]      = fmaxf(c0[g] + b1l, 0.f);
    lds_u[wave][r * 33 + 16 + l16] = fmaxf(c1[g] + b1h, 0.f);
  }
  __builtin_amdgcn_wave_barrier();  // per-wave DS ops are in-order

  // ---- GEMM2: V = U@W2 + B2, A from LDS (lane = row), B = staged W2 ----
  v8f d0 = {}, d1 = {};
#pragma unroll
  for (int kb = 0; kb < 32; kb += 4) {
    const int k = kb + 2 * half;
    const float* w2l = lds_w + 1024;
    v2f a;  a.x = lds_u[wave][l16 * 33 + k]; a.y = lds_u[wave][l16 * 33 + k + 1];
    v2f bl; bl.x = w2l[k * 32 + l16];        bl.y = w2l[(k + 1) * 32 + l16];
    v2f bh; bh.x = w2l[k * 32 + 16 + l16];   bh.y = w2l[(k + 1) * 32 + 16 + l16];
    d0 = wmma_f32_4(a, bl, d0);
    d1 = wmma_f32_4(a, bh, d1);
  }

  // ---- bias + relu + store + BN partial sums (column = lane in C layout) ----
  const float b2l = B2[l16], b2h = B2[16 + l16];
  float s0 = 0.f, s1 = 0.f, q0 = 0.f, q1 = 0.f;
#pragma unroll
  for (int g = 0; g < 8; ++g) {
    const int r = row0 + g + 8 * half;
    float v0 = fmaxf(d0[g] + b2l, 0.f);
    float v1 = fmaxf(d1[g] + b2h, 0.f);
    if (r >= n_valid) { v0 = 0.f; v1 = 0.f; }  // keep padded tail zero
    R[(size_t)r * 32 + l16]      = v0;
    R[(size_t)r * 32 + 16 + l16] = v1;
    s0 += v0; s1 += v1; q0 += v0 * v0; q1 += v1 * v1;
  }
  // lane L and L+16 hold the same column (rows g vs g+8): fold halves.
  s0 += __shfl_xor(s0, 16, 32);
  s1 += __shfl_xor(s1, 16, 32);
  q0 += __shfl_xor(q0, 16, 32);
  q1 += __shfl_xor(q1, 16, 32);
  if (half == 0) {
    lds_red[wave][l16]      = s0;  // sum, cols 0..15
    lds_red[wave][16 + l16] = s1;  // sum, cols 16..31
    lds_red[wave][32 + l16] = q0;  // sumsq, cols 0..15
    lds_red[wave][48 + l16] = q1;  // sumsq, cols 16..31
  }
  __syncthreads();
  if (tid < 64) {
    float acc = 0.f;
#pragma unroll
    for (int w = 0; w < 8; ++w) acc += lds_red[w][tid];
    atomicAdd(&bns[tid], acc);  // bns[0..31]=sum, [32..63]=sumsq
  }
}

// Turn BN batch stats into the per-feature affine: sc = g*rsqrt(var+eps),
// sh = beta - mu*sc.
__global__ void k_affine(const float* __restrict__ bns,
                         const float* __restrict__ gamma,
                         const float* __restrict__ beta,
                         float* __restrict__ aff, float inv_n) {
  int f = threadIdx.x;
  if (f >= 32) return;
  float mu = bns[f] * inv_n;
  float var = bns[32 + f] * inv_n - mu * mu;
  float sc = gamma[f] * rsqrtf(var + 1e-5f);
  aff[f] = sc;
  aff[32 + f] = beta[f] - mu * sc;
}

__global__ __launch_bounds__(256) void k_zero(float* __restrict__ p, long n) {
  long i = (long)blockIdx.x * 256 + threadIdx.x;
  if (i < n) p[i] = 0.f;
}

// Mean-pool numerators/denominators via atomics; BN affine applied on read.
__global__ __launch_bounds__(256) void k_pool(const float* __restrict__ H,
                                              const int* __restrict__ batch,
                                              const float* __restrict__ aff,
                                              float* __restrict__ gsum,
                                              float* __restrict__ gcnt,
                                              long total) {
  long i = (long)blockIdx.x * 256 + threadIdx.x;
  if (i >= total) return;
  int n = (int)(i >> 5), f = (int)(i & 31);
  int g = batch[n];
  atomicAdd(&gsum[(size_t)g * 32 + f], fmaf(H[i], aff[f], aff[32 + f]));
  if (f == 0) atomicAdd(&gcnt[g], 1.f);
}

// Head: emb = sums/max(cnt,1); z = relu(emb@wb+bb); out = sigmoid(z@wm+bm).
__global__ __launch_bounds__(256) void k_head(const float* __restrict__ gsum,
                                              const float* __restrict__ gcnt,
                                              const float* __restrict__ wb,
                                              const float* __restrict__ bb,
                                              const float* __restrict__ wm,
                                              const float* __restrict__ bm,
                                              float* __restrict__ out, int ng) {
  int g = blockIdx.x * 256 + threadIdx.x;
  if (g >= ng) return;
  float inv = 1.f / fmaxf(gcnt[g], 1.f);
  float emb[32];
#pragma unroll
  for (int k = 0; k < 32; ++k) emb[k] = gsum[(size_t)g * 32 + k] * inv;
  float acc = bm[0];
#pragma unroll
  for (int j = 0; j < 16; ++j) {
    float z = bb[j];
#pragma unroll
    for (int k = 0; k < 32; ++k) z += emb[k] * wb[k * 16 + j];
    acc += fmaxf(z, 0.f) * wm[j];
  }
  out[g] = 1.f / (1.f + __expf(-acc));
}

// ---------------------------------------------------------------------------
extern "C" void kernel_launch(void* const* d_in, const int* in_sizes, int n_in,
                              void* d_out, int out_size, void* d_ws,
                              size_t ws_size, hipStream_t stream) {
  const float* x     = (const float*)d_in[0];
  const int*   ei    = (const int*)d_in[1];
  const int*   batch = (const int*)d_in[2];
  const float *w1a = (const float*)d_in[3],  *b1a = (const float*)d_in[4];
  const float *w1b = (const float*)d_in[5],  *b1b = (const float*)d_in[6];
  const float *g1  = (const float*)d_in[7],  *be1 = (const float*)d_in[8];
  const float *w2a = (const float*)d_in[9],  *b2a = (const float*)d_in[10];
  const float *w2b = (const float*)d_in[11], *b2b = (const float*)d_in[12];
  const float *g2  = (const float*)d_in[13], *be2 = (const float*)d_in[14];
  const float *w3a = (const float*)d_in[15], *b3a = (const float*)d_in[16];
  const float *w3b = (const float*)d_in[17], *b3b = (const float*)d_in[18];
  const float *g3  = (const float*)d_in[19], *be3 = (const float*)d_in[20];
  const float *wb  = (const float*)d_in[21], *bb  = (const float*)d_in[22];
  const float *wm  = (const float*)d_in[23], *bm  = (const float*)d_in[24];

  const int N_ = in_sizes[0] / 6;     // 150000
  const int E_ = in_sizes[1] / 2;     // 2400000
  const int G_ = out_size;            // 1024
  const int Np = (N_ + 127) & ~127;   // 128-aligned row count
  const size_t np32 = (size_t)Np * 32;

  float* ws   = (float*)d_ws;
  float* BX   = ws;                  // padded x (layer 1 source)
  float* BT   = ws + np32;           // t = agg + h (GIN MLP input)
  float* BR   = ws + 2 * np32;       // pre-BN layer output
  float* W1P  = ws + 3 * np32;       // [1024] padded w1a
  float* BNS  = W1P + 1024;          // [64] BN sum / sumsq
  float* AFF  = BNS + 64;            // [64] sc[32] | sh[32]
  float* GSUM = AFF + 64;            // [G,32]
  float* GCNT = GSUM + (size_t)G_ * 32;  // [G]

  const dim3 blk(256);
  const long totalP = (long)np32;
  const int gridP = (int)((totalP + 255) / 256);
  const int gridE = (int)(((long)E_ * 32 + 255) / 256);
  const int gridM = Np / 128;  // 8 wave-tiles (16 rows each) per block
  const float inv_n = 1.f / (float)N_;

  k_pad_w1<<<5, blk, 0, stream>>>(w1a, W1P, AFF);  // weights + identity affine

  // ---- layer 1: src BX (padded x, identity affine) ----
  k_init_layer1<<<gridP, blk, 0, stream>>>(x, BX, BT, BNS, N_, totalP);
  k_edges<<<gridE, blk, 0, stream>>>(ei, BX, AFF, BT, E_);
  k_gin_mlp<<<gridM, blk, 0, stream>>>(BT, W1P, b1a, w1b, b1b, BR, BNS, N_);
  k_affine<<<1, 32, 0, stream>>>(BNS, g1, be1, AFF, inv_n);

  // ---- layer 2: src BR with layer-1 affine ----
  k_init_aff<<<gridP, blk, 0, stream>>>(BR, AFF, BT, BNS, totalP);
  k_edges<<<gridE, blk, 0, stream>>>(ei, BR, AFF, BT, E_);
  k_gin_mlp<<<gridM, blk, 0, stream>>>(BT, w2a, b2a, w2b, b2b, BR, BNS, N_);
  k_affine<<<1, 32, 0, stream>>>(BNS, g2, be2, AFF, inv_n);

  // ---- layer 3: src BR with layer-2 affine ----
  k_init_aff<<<gridP, blk, 0, stream>>>(BR, AFF, BT, BNS, totalP);
  k_edges<<<gridE, blk, 0, stream>>>(ei, BR, AFF, BT, E_);
  k_gin_mlp<<<gridM, blk, 0, stream>>>(BT, w3a, b3a, w3b, b3b, BR, BNS, N_);
  k_affine<<<1, 32, 0, stream>>>(BNS, g3, be3, AFF, inv_n);

  // ---- readout: pool applies layer-3 affine on the fly ----
  const long poolTot = (long)G_ * 32 + G_;  // GSUM + GCNT contiguous
  k_zero<<<(int)((poolTot + 255) / 256), blk, 0, stream>>>(GSUM, poolTot);
  k_pool<<<(int)(((long)N_ * 32 + 255) / 256), blk, 0, stream>>>(
      BR, batch, AFF, GSUM, GCNT, (long)N_ * 32);
  k_head<<<(G_ + 255) / 256, blk, 0, stream>>>(GSUM, GCNT, wb, bb, wm, bm,
                                               (float*)d_out, G_);
}